// DenseMLPQMatrixDecoder_45011257262439
// MI455X (gfx1250) — compile-verified
//
#include <hip/hip_runtime.h>

// CDNA5 / gfx1250. wave32. WMMA f32 16x16x4 for the MLP GEMMs;
// per-lane 4x4 expm (scaling & squaring); float4 streaming broadcast.

typedef __attribute__((ext_vector_type(2))) float v2f;
typedef __attribute__((ext_vector_type(8))) float v8f;

#define V_TOK   1024
#define S_SITES 1024
#define NA      4
#define T_INF_F 1000.0f

// ---------------------------------------------------------------------------
// Kernel 1: MLP  emb[16x8] -> relu -> [16x16] -> relu -> [16x16] -> logq
// One wave (32 threads) per 16-row tile of V. Uses V_WMMA_F32_16X16X4_F32.
//
// A-fragment layout (16x4 f32): lane = m + 16*hi holds A[m][2*hi + j] in reg j.
// B-fragment layout (4x16 f32): lane = n + 16*hi holds B[2*hi + j][n] in reg j.
// C/D layout (16x16 f32):       lane = n + 16*hi, reg i = D[i + 8*hi][n].
// ---------------------------------------------------------------------------
__global__ __launch_bounds__(32) void dmlp_mlp_wmma_kernel(
    const float* __restrict__ emb,  const float* __restrict__ W0,
    const float* __restrict__ b0,   const float* __restrict__ W1,
    const float* __restrict__ b1,   const float* __restrict__ Wout,
    const float* __restrict__ bout, float* __restrict__ logq)
{
    __shared__ float hbuf[16 * 16];
    const int l   = threadIdx.x;
    const int row = l & 15;      // n (or m for A-frags)
    const int hi  = l >> 4;      // which K/M half this lane covers
    const int v0  = blockIdx.x * 16;

    // ---- Layer 0: [16x8] @ [8x16], K-chunks of 4 ----
    v8f c0 = {};
#pragma unroll
    for (int kk = 0; kk < 2; ++kk) {
        const int k0 = 4 * kk + 2 * hi;
        v2f a, b;
        a.x = emb[(v0 + row) * 8 + k0 + 0];
        a.y = emb[(v0 + row) * 8 + k0 + 1];
        b.x = W0[(k0 + 0) * 16 + row];
        b.y = W0[(k0 + 1) * 16 + row];
        c0 = __builtin_amdgcn_wmma_f32_16x16x4_f32(false, a, false, b,
                                                   (short)0, c0, false, false);
    }
    // bias + ReLU, stage h to LDS in [m][n] order
#pragma unroll
    for (int i = 0; i < 8; ++i) {
        float x = c0[i] + b0[row];
        hbuf[(i + 8 * hi) * 16 + row] = x > 0.0f ? x : 0.0f;
    }
    __syncthreads();

    // ---- Layer 1: [16x16] @ [16x16] ----
    v8f c1 = {};
#pragma unroll
    for (int kk = 0; kk < 4; ++kk) {
        const int k0 = 4 * kk + 2 * hi;
        v2f a, b;
        a.x = hbuf[row * 16 + k0 + 0];
        a.y = hbuf[row * 16 + k0 + 1];
        b.x = W1[(k0 + 0) * 16 + row];
        b.y = W1[(k0 + 1) * 16 + row];
        c1 = __builtin_amdgcn_wmma_f32_16x16x4_f32(false, a, false, b,
                                                   (short)0, c1, false, false);
    }
    __syncthreads();
#pragma unroll
    for (int i = 0; i < 8; ++i) {
        float x = c1[i] + b1[row];
        hbuf[(i + 8 * hi) * 16 + row] = x > 0.0f ? x : 0.0f;
    }
    __syncthreads();

    // ---- Output layer: [16x16] @ [16x16] -> log_q ----
    v8f c2 = {};
#pragma unroll
    for (int kk = 0; kk < 4; ++kk) {
        const int k0 = 4 * kk + 2 * hi;
        v2f a, b;
        a.x = hbuf[row * 16 + k0 + 0];
        a.y = hbuf[row * 16 + k0 + 1];
        b.x = Wout[(k0 + 0) * 16 + row];
        b.y = Wout[(k0 + 1) * 16 + row];
        c2 = __builtin_amdgcn_wmma_f32_16x16x4_f32(false, a, false, b,
                                                   (short)0, c2, false, false);
    }
#pragma unroll
    for (int i = 0; i < 8; ++i)
        logq[(v0 + i + 8 * hi) * 16 + row] = c2[i] + bout[row];
}

// ---------------------------------------------------------------------------
// Kernel 2: per-v Q build + expm(1000*Q) via scaling & squaring, one lane per v.
// ||1000*Q||_inf == 2000 exactly (diag -1000, off-diag row sum 1000), so with
// s = 14, ||X|| <= 0.1221; Taylor to order 8 leaves < 1e-12 truncation error.
// ---------------------------------------------------------------------------
__device__ __forceinline__ void dmlp_mm4(const float* __restrict__ X,
                                         const float* __restrict__ Y,
                                         float* __restrict__ Z)
{
#pragma unroll
    for (int i = 0; i < 4; ++i)
#pragma unroll
        for (int j = 0; j < 4; ++j) {
            float s = 0.0f;
#pragma unroll
            for (int k = 0; k < 4; ++k)
                s = fmaf(X[i * 4 + k], Y[k * 4 + j], s);
            Z[i * 4 + j] = s;
        }
}

__global__ __launch_bounds__(256) void dmlp_qexpm_kernel(
    const float* __restrict__ logq, float* __restrict__ probs)
{
    const int v = blockIdx.x * blockDim.x + threadIdx.x;
    if (v >= V_TOK) return;

    float lq[16];
#pragma unroll
    for (int i = 0; i < 16; ++i) lq[i] = logq[v * 16 + i];

    // Build generator Q: exp off-diagonal, row-normalize, diag = -1
    float Q[16];
#pragma unroll
    for (int i = 0; i < 4; ++i) {
        float rs = 0.0f;
#pragma unroll
        for (int j = 0; j < 4; ++j)
            if (j != i) { float e = __expf(lq[i * 4 + j]); Q[i * 4 + j] = e; rs += e; }
        const float inv = 1.0f / rs;
#pragma unroll
        for (int j = 0; j < 4; ++j)
            Q[i * 4 + j] = (j == i) ? -1.0f : Q[i * 4 + j] * inv;
    }

    // X = (T_INF * Q) / 2^14
    const float scale = T_INF_F * (1.0f / 16384.0f);
    float X[16], P[16], T[16], U[16];
#pragma unroll
    for (int i = 0; i < 16; ++i) {
        X[i] = Q[i] * scale;
        T[i] = X[i];
        P[i] = X[i] + ((i % 5 == 0) ? 1.0f : 0.0f);  // I + X
    }
    // Taylor terms k = 2..8
#pragma unroll
    for (int k = 2; k <= 8; ++k) {
        dmlp_mm4(T, X, U);
        const float invk = 1.0f / (float)k;
#pragma unroll
        for (int i = 0; i < 16; ++i) { T[i] = U[i] * invk; P[i] += T[i]; }
    }
    // 14 squarings
#pragma unroll
    for (int r = 0; r < 14; ++r) {
        dmlp_mm4(P, P, U);
#pragma unroll
        for (int i = 0; i < 16; ++i) P[i] = U[i];
    }
    // first row = stationary probs
#pragma unroll
    for (int j = 0; j < 4; ++j) probs[v * 4 + j] = P[j];
}

// ---------------------------------------------------------------------------
// Kernel 3: broadcast probs[v][0:4] across S sites. Bandwidth-bound: 16 MB of
// float4 (global_store_b128) coalesced stores. One block per v.
// ---------------------------------------------------------------------------
__global__ __launch_bounds__(256) void dmlp_bcast_kernel(
    const float* __restrict__ probs, float4* __restrict__ out)
{
    const int v = blockIdx.x;
    const float4 pv = ((const float4*)probs)[v];
    float4* o = out + (size_t)v * S_SITES;
    const int t = threadIdx.x;
#pragma unroll
    for (int r = 0; r < 4; ++r)
        o[t + r * 256] = pv;
}

// ---------------------------------------------------------------------------
extern "C" void kernel_launch(void* const* d_in, const int* in_sizes, int n_in,
                              void* d_out, int out_size, void* d_ws, size_t ws_size,
                              hipStream_t stream)
{
    const float* emb  = (const float*)d_in[0];  // [V,8]
    // d_in[1] = site_positions [S,4] : unused (Q identical across sites)
    const float* W0   = (const float*)d_in[2];  // [8,16]
    const float* b0   = (const float*)d_in[3];  // [16]
    const float* W1   = (const float*)d_in[4];  // [16,16]
    const float* b1   = (const float*)d_in[5];  // [16]
    const float* Wout = (const float*)d_in[6];  // [16,16]
    const float* bout = (const float*)d_in[7];  // [16]
    float* out   = (float*)d_out;               // [V,S,4]

    float* logq  = (float*)d_ws;                // V*16 floats = 64 KB
    float* probs = logq + V_TOK * 16;           // V*4  floats = 16 KB

    dmlp_mlp_wmma_kernel<<<V_TOK / 16, 32, 0, stream>>>(
        emb, W0, b0, W1, b1, Wout, bout, logq);
    dmlp_qexpm_kernel<<<V_TOK / 256, 256, 0, stream>>>(logq, probs);
    dmlp_bcast_kernel<<<V_TOK, 256, 0, stream>>>(probs, (float4*)out);
}